// RGCN_64132451664513
// MI455X (gfx1250) — compile-verified
//
#include <hip/hip_runtime.h>

typedef float v2f __attribute__((ext_vector_type(2)));
typedef float v4f __attribute__((ext_vector_type(4)));
typedef float v8f __attribute__((ext_vector_type(8)));

__device__ __forceinline__ void atomic_add_f32(float* p, float v) {
    __hip_atomic_fetch_add(p, v, __ATOMIC_RELAXED, __HIP_MEMORY_SCOPE_AGENT);
}

// ---------------------------------------------------------------------------
// Zero-fill workspace regions (grid-stride).
// ---------------------------------------------------------------------------
__global__ void rgcn_zero_f32(float* __restrict__ p, size_t n) {
    size_t i = (size_t)blockIdx.x * blockDim.x + threadIdx.x;
    size_t stride = (size_t)gridDim.x * blockDim.x;
    for (; i < n; i += stride) p[i] = 0.0f;
}

// ---------------------------------------------------------------------------
// counts[et*N + src] += 1  (one thread per triple). triples: (T,3) int32.
// ---------------------------------------------------------------------------
__global__ void rgcn_degree_count(const int* __restrict__ triples,
                                  float* __restrict__ counts,
                                  int nT, int nNodes) {
    int e = blockIdx.x * blockDim.x + threadIdx.x;
    if (e >= nT) return;
    int s = triples[3 * e + 0];
    int r = triples[3 * e + 1];
    atomic_add_f32(&counts[(size_t)r * nNodes + s], 1.0f);
}

// ---------------------------------------------------------------------------
// af[row, f..f+3] += x[dst, f..f+3] / counts[row],  row = et*N + src.
// 16 threads per triple, float4 per thread: 256B coalesced gather per edge,
// 4 float atomics per thread. af = R*N*64*4B = 89.6MB < 192MB L2, so the
// atomic traffic is absorbed by the L2 atomic units.
// ---------------------------------------------------------------------------
__global__ void rgcn_scatter(const int* __restrict__ triples,
                             const float* __restrict__ counts,
                             const float* __restrict__ xin,
                             float* __restrict__ af,
                             int nT, int nNodes) {
    long long t = (long long)blockIdx.x * blockDim.x + threadIdx.x;
    int e  = (int)(t >> 4);
    int f0 = (int)(t & 15) * 4;
    if (e >= nT) return;
    int s = triples[3 * e + 0];
    int r = triples[3 * e + 1];
    int d = triples[3 * e + 2];
    size_t row = (size_t)r * nNodes + s;
    float inv = 1.0f / counts[row];
    v4f xv = *(const v4f*)(xin + (size_t)d * 64 + f0);
    float* dst = af + row * 64 + f0;
    atomic_add_f32(dst + 0, inv * xv.x);
    atomic_add_f32(dst + 1, inv * xv.y);
    atomic_add_f32(dst + 2, inv * xv.z);
    atomic_add_f32(dst + 3, inv * xv.w);
}

// ---------------------------------------------------------------------------
// out[n, o] = sum_k af_stacked[n, k] * w[k, o] + bias[o]  (optionally ReLU)
//   K = nRels*64, af_stacked[n, r*64+i] = af[(r*N + n)*64 + i]
//   w is (R, 64, 64) row-major == already the stacked (K, 64) B matrix.
//
// One wave -> one 16x16 C tile via V_WMMA_F32_16X16X4_F32.
// Block = 256 threads = 8 waves = 32 nodes x 64 output cols.
//
// ISA VGPR layouts (CDNA5 ISA):
//   A 16x4  : lanes 0-15 row=lane, {v0,v1}={K0,K1}; lanes 16-31 {K2,K3}
//   B 4x16  : lanes 0-15 col=lane, {v0,v1}={K0,K1}; lanes 16-31 {K2,K3}
//   C 16x16 : col = lane&15, row = v + 8*(lane>=16)
//
// Tail handling: C row m depends only on A row m, so clamping out-of-range
// node indices (instead of zeroing) is safe -- those C rows are never stored.
// This keeps EXEC all-ones with no divergence in the hot loop (WMMA requires
// EXEC==all 1s anyway).
// ---------------------------------------------------------------------------
__global__ void rgcn_transform(const float* __restrict__ af,
                               const float* __restrict__ w,
                               const float* __restrict__ bias,
                               float* __restrict__ out,
                               int nNodes, int nRels, int doRelu) {
    int wave = threadIdx.x >> 5;
    int lane = threadIdx.x & 31;
    int half = lane >> 4;     // 0: lanes 0-15, 1: lanes 16-31
    int lm   = lane & 15;

    int rowBase = blockIdx.x * 32 + (wave >> 2) * 16;  // node tile
    int colBase = (wave & 3) * 16;                     // output-column tile

    int m = rowBase + lm;
    if (m >= nNodes) m = nNodes - 1;   // clamp: affects only unstored C rows

    v8f c = {0.f, 0.f, 0.f, 0.f, 0.f, 0.f, 0.f, 0.f};

    for (int r = 0; r < nRels; ++r) {
        // This lane-half's first K element within relation r is i = half*2.
        const float* ap = af + ((size_t)r * nNodes + m) * 64 + half * 2;
        const float* bp = w  + (size_t)r * 64 * 64 + (half * 2) * 64 + colBase + lm;
#pragma unroll
        for (int ii = 0; ii < 16; ++ii) {
            v2f a;
            a.x = ap[ii * 4 + 0];
            a.y = ap[ii * 4 + 1];
            v2f b;
            b.x = bp[(ii * 4 + 0) * 64];
            b.y = bp[(ii * 4 + 1) * 64];
            // 8 args: (neg_a, A, neg_b, B, c_mod, C, reuse_a, reuse_b)
            c = __builtin_amdgcn_wmma_f32_16x16x4_f32(false, a, false, b,
                                                      (short)0, c, false, false);
        }
    }

    int col = colBase + lm;
    float bv = bias[col];
#pragma unroll
    for (int v = 0; v < 8; ++v) {
        int row = rowBase + v + 8 * half;
        if (row < nNodes) {
            float val = c[v] + bv;
            if (doRelu) val = fmaxf(val, 0.f);
            out[(size_t)row * 64 + col] = val;
        }
    }
}

// ---------------------------------------------------------------------------
// Orchestration. Inputs (setup_inputs order):
//   0: x (N*64 f32)  1: w1 (R*64*64 f32)  2: b1 (64 f32)
//   3: w2 (R*64*64)  4: b2 (64)           5: triples (T*3 int32)
//   6: num_nodes     7: num_edge_types
// Workspace: counts (R*N f32) | af (R*N*64 f32) | h (N*64 f32) ~= 108 MB.
// ---------------------------------------------------------------------------
extern "C" void kernel_launch(void* const* d_in, const int* in_sizes, int n_in,
                              void* d_out, int out_size, void* d_ws, size_t ws_size,
                              hipStream_t stream) {
    const float* x   = (const float*)d_in[0];
    const float* w1  = (const float*)d_in[1];
    const float* b1  = (const float*)d_in[2];
    const float* w2  = (const float*)d_in[3];
    const float* b2  = (const float*)d_in[4];
    const int*   tri = (const int*)d_in[5];

    int nNodes = in_sizes[0] / 64;
    int nRels  = in_sizes[1] / (64 * 64);
    int nT     = in_sizes[5] / 3;

    size_t cN  = (size_t)nRels * nNodes;   // counts elements
    size_t afN = cN * 64;                  // af elements

    float* counts = (float*)d_ws;
    float* af     = counts + cN;
    float* h      = af + afN;

    const int threads = 256;
    int countBlocks   = (nT + threads - 1) / threads;
    int scatterBlocks = (int)(((size_t)nT * 16 + threads - 1) / threads);
    int gemmBlocks    = (nNodes + 31) / 32;

    // Degree counts (shared by both layers: rows don't change).
    rgcn_zero_f32<<<2048, threads, 0, stream>>>(counts, cN);
    rgcn_degree_count<<<countBlocks, threads, 0, stream>>>(tri, counts, nT, nNodes);

    // ---- Layer 1: aggregate x -> af, transform (w1,b1) + ReLU -> h ----
    rgcn_zero_f32<<<4096, threads, 0, stream>>>(af, afN);
    rgcn_scatter<<<scatterBlocks, threads, 0, stream>>>(tri, counts, x, af, nT, nNodes);
    rgcn_transform<<<gemmBlocks, threads, 0, stream>>>(af, w1, b1, h, nNodes, nRels, 1);

    // ---- Layer 2: aggregate h -> af, transform (w2,b2) -> out ----
    rgcn_zero_f32<<<4096, threads, 0, stream>>>(af, afN);
    rgcn_scatter<<<scatterBlocks, threads, 0, stream>>>(tri, counts, h, af, nT, nNodes);
    rgcn_transform<<<gemmBlocks, threads, 0, stream>>>(af, w2, b2, (float*)d_out,
                                                      nNodes, nRels, 0);
}